// NN_17_71983651881627
// MI455X (gfx1250) — compile-verified
//
#include <hip/hip_runtime.h>
#include <stdint.h>

#define T_STEPS 64
#define BATCH   65536
#define BLOCK   256

struct KParams {
  const float* inputs;
  const float* S0;  const float* D0;  const float* sinc_w;
  const float* rw_w1;  const float* rw_b1;  const float* rw_w2;  const float* rw_b2;
  const float* nd_w1;  const float* nd_b1;  const float* nd_w2;  const float* nd_b2;
  const float* pls_w1; const float* pls_b1; const float* pls_w2; const float* pls_b2;
  const float* sc_w1;  const float* sc_b1;  const float* sc_w2;  const float* sc_b2;
  const float* bs_w1;  const float* bs_b1;  const float* bs_w2;  const float* bs_b2;
  float* out;
};

// Fast hardware transcendentals: v_exp_f32 is exp2, v_log_f32 is log2.
__device__ __forceinline__ float fexp(float x){ return __builtin_amdgcn_exp2f(x * 1.4426950408889634f); }
__device__ __forceinline__ float flog(float x){ return __builtin_amdgcn_logf(x) * 0.6931471805599453f; }
__device__ __forceinline__ float frcp(float x){ return __builtin_amdgcn_rcpf(x); }
__device__ __forceinline__ float clampf(float x, float lo, float hi){ return fminf(fmaxf(x, lo), hi); }

// jax.nn.softplus(x) = max(x,0) + log1p(exp(-|x|))  (stable form)
__device__ __forceinline__ float softplus_f(float x){
  return fmaxf(x, 0.0f) + flog(1.0f + fexp(-fabsf(x)));
}
// mish(z) = z*tanh(softplus(z)) = z*((1+e^z)^2-1)/((1+e^z)^2+1): 1 exp + 1 rcp
__device__ __forceinline__ float mish_f(float z){
  float p  = fexp(fminf(z, 30.0f));   // clamp avoids overflow; ratio saturates to 1
  float q  = 1.0f + p;
  float q2 = q * q;
  return z * (q2 - 1.0f) * frcp(q2 + 1.0f);
}
__device__ __forceinline__ float sigmoid_f(float x){ return frcp(1.0f + fexp(-x)); }

// Async global->LDS copy of 3 consecutive dwords (ASYNCcnt-tracked).
// INST_OFFSET applies to both the LDS destination and the global address,
// so one VGPR pair of operands covers all three dwords.
__device__ __forceinline__ void async_load3(unsigned lds_off, const float* g){
  asm volatile(
    "global_load_async_to_lds_b32 %0, %1, off\n\t"
    "global_load_async_to_lds_b32 %0, %1, off offset:4\n\t"
    "global_load_async_to_lds_b32 %0, %1, off offset:8"
    :
    : "v"(lds_off), "v"(g)
    : "memory");
}
// Generic pointers into LDS carry the aperture in the high 32 bits; the low
// 32 bits ARE the LDS byte offset (ISA 10.2 aperture mapping).
__device__ __forceinline__ unsigned lds_off_of(const void* p){
  return (unsigned)(size_t)p;
}

__global__ __launch_bounds__(BLOCK) void fsrs_scan(KParams P){
  __shared__ float xb[2][BLOCK * 3];   // double-buffered per-lane inputs (12 B/lane)
  const int tid = threadIdx.x;
  const int b   = blockIdx.x * BLOCK + tid;

  // Uniform scalar parameters (scalar-cache loads).
  const float S00=P.S0[0], S01=P.S0[1], S02=P.S0[2], S03=P.S0[3];
  const float D00=P.D0[0], D01=P.D0[1], D02=P.D0[2], D03=P.D0[3];
  const float sw0=P.sinc_w[0], sw1=P.sinc_w[1], sw2=P.sinc_w[2];
  const float rww0=P.rw_w1[0], rww1=P.rw_w1[1], rww2=P.rw_w1[2];
  const float rwb1=P.rw_b1[0], rww2o=P.rw_w2[0], rwb2=P.rw_b2[0];
  const float ndw0=P.nd_w1[0], ndw1=P.nd_w1[1], ndw2=P.nd_w1[2];
  const float ndb1=P.nd_b1[0], ndw2o=P.nd_w2[0], ndb2=P.nd_b2[0];
  const float plw0=P.pls_w1[0], plw1=P.pls_w1[1];
  const float plb1=P.pls_b1[0], plw2o=P.pls_w2[0], plb2=P.pls_b2[0];
  const float scw0=P.sc_w1[0], scw1=P.sc_w1[1], scw2=P.sc_w1[2];
  const float scb1=P.sc_b1[0], scw2o=P.sc_w2[0], scb2=P.sc_b2[0];
  const float bsw0=P.bs_w1[0], bsw1=P.bs_w1[1];
  const float bsb1=P.bs_b1[0], bsw2o=P.bs_w2[0], bsb2=P.bs_b2[0];
  const float ew0 = fexp(sw0);         // exp(sinc_w[0]) hoisted out of the loop

  const float* gbase = P.inputs + (size_t)b * 3;
  const unsigned lds0 = lds_off_of(&xb[0][3*tid]);
  const unsigned lds1 = lds_off_of(&xb[1][3*tid]);

  // Prime the pipeline: t=0 into buffer 0, t=1 into buffer 1.
  async_load3(lds0, gbase);
  async_load3(lds1, gbase + (size_t)BATCH * 3);

  // ---- Peeled init step (t == 0): state0 == ones -> is_init is true, the
  // reference computes the heavy chain but discards it via jnp.where. Output
  // is just the S0/D0 table lookup on the rating.
  float s, d;
  {
    asm volatile("s_wait_asynccnt 3" ::: "memory");  // buffer 0 complete
    const float rating = xb[0][3*tid + 1];
    const int  r_int = (int)rating;
    const bool valid = (r_int >= 1) && (r_int <= 4);
    const int  idx   = r_int < 1 ? 0 : (r_int > 4 ? 3 : r_int - 1);
    float s_init = (idx==0) ? S00 : (idx==1) ? S01 : (idx==2) ? S02 : S03;
    float d_init = (idx==0) ? D00 : (idx==1) ? D01 : (idx==2) ? D02 : D03;
    if (!valid) { s_init = 0.0f; d_init = 0.0f; }
    s = clampf(s_init, 0.01f, 36500.0f);
    d = d_init;
    float2 o; o.x = s; o.y = d;
    *(float2*)(P.out + (size_t)b * 2) = o;
  }

  // ---- Steady-state scan, t = 1 .. 63 (branchless prefetch; the final
  // iteration redundantly re-fetches t=63 into the dead buffer, which is
  // covered by the implicit wait-idle at s_endpgm).
  #pragma unroll 1
  for (int t = 1; t < T_STEPS; ++t) {
    const int cur = t & 1;
    const int tn  = (t + 1 < T_STEPS) ? (t + 1) : (T_STEPS - 1);
    async_load3(cur ? lds0 : lds1, gbase + (size_t)tn * (size_t)BATCH * 3);
    // <=3 outstanding => the 3 loads for the current buffer (issued last
    // iteration) have completed, since async loads retire in order.
    asm volatile("s_wait_asynccnt 3" ::: "memory");

    const float dt     = xb[cur][3*tid + 0];
    const float rating = xb[cur][3*tid + 1];
    const float lapses = xb[cur][3*tid + 2];

    // rt = clip(0.9^(dt/s), 1e-4, 0.9999);  0.9^x = exp2(log2(0.9)*x)
    const float rt = clampf(__builtin_amdgcn_exp2f(-0.15200309344504997f * dt * frcp(s)),
                            1e-4f, 0.9999f);
    // rw MLP over [d, s, rt]
    const float rw_h   = mish_f(d*rww0 + s*rww1 + rt*rww2 + rwb1);
    const float rw_pre = softplus_f(rww2o*rw_h + rwb2);
    const float rw     = clampf(fexp(-rw_pre), 1e-4f, 0.9999f);
    // sr = clip(LOG09/log(rw)*dt, 0.01, 36500)
    const float sr     = clampf(-0.10536051565782628f * frcp(flog(rw)) * dt, 0.01f, 36500.0f);
    // new difficulty MLP over [d, rw, rating]
    const float nd_h    = mish_f(d*ndw0 + rw*ndw1 + rating*ndw2 + ndb1);
    const float new_d   = sigmoid_f(ndw2o*nd_h + ndb2);
    // post-lapse stability MLP over [rw, lapses]
    const float pl_h = mish_f(rw*plw0 + lapses*plw1 + plb1);
    const float pls  = clampf(softplus_f(plw2o*pl_h + plb2), 0.01f, 36500.0f);
    // theoretical stability increase: sr^(-sw1) = exp2(-sw1*log2(sr)), sr>0
    const float srp    = __builtin_amdgcn_exp2f(-sw1 * __builtin_amdgcn_logf(sr));
    const float sinc_t = 1.0f + ew0 * (5.0f*(1.0f - new_d) + 1.0f) * srp * fexp(-rw*sw2);
    // nn stability increase MLP over [new_d, sr, rw]
    const float sc_h    = mish_f(new_d*scw0 + sr*scw1 + rw*scw2 + scb1);
    const float sinc_nn = 1.0f + softplus_f(scw2o*sc_h + scb2);
    // best stability increase MLP over [sinc_t, sinc_nn]
    const float bs_h      = mish_f(sinc_t*bsw0 + sinc_nn*bsw1 + bsb1);
    const float best_sinc = 1.0f + softplus_f(bsw2o*bs_h + bsb2);

    const float s_else = (rating > 1.0f) ? sr * best_sinc : pls;
    const float new_s  = clampf(s_else, 0.01f, 36500.0f);

    float2 o; o.x = new_s; o.y = new_d;             // 8B-aligned -> global_store_b64
    *(float2*)(P.out + ((size_t)t * BATCH + (size_t)b) * 2) = o;
    s = new_s; d = new_d;
  }

  // final_state appended after outputs
  float2 fs; fs.x = s; fs.y = d;
  *(float2*)(P.out + (size_t)T_STEPS * BATCH * 2 + (size_t)b * 2) = fs;
}

extern "C" void kernel_launch(void* const* d_in, const int* in_sizes, int n_in,
                              void* d_out, int out_size, void* d_ws, size_t ws_size,
                              hipStream_t stream) {
  (void)in_sizes; (void)n_in; (void)out_size; (void)d_ws; (void)ws_size;
  KParams P;
  P.inputs = (const float*)d_in[0];
  P.S0     = (const float*)d_in[1];
  P.D0     = (const float*)d_in[2];
  P.sinc_w = (const float*)d_in[3];
  P.rw_w1  = (const float*)d_in[4];  P.rw_b1  = (const float*)d_in[5];
  P.rw_w2  = (const float*)d_in[6];  P.rw_b2  = (const float*)d_in[7];
  P.nd_w1  = (const float*)d_in[8];  P.nd_b1  = (const float*)d_in[9];
  P.nd_w2  = (const float*)d_in[10]; P.nd_b2  = (const float*)d_in[11];
  P.pls_w1 = (const float*)d_in[12]; P.pls_b1 = (const float*)d_in[13];
  P.pls_w2 = (const float*)d_in[14]; P.pls_b2 = (const float*)d_in[15];
  P.sc_w1  = (const float*)d_in[16]; P.sc_b1  = (const float*)d_in[17];
  P.sc_w2  = (const float*)d_in[18]; P.sc_b2  = (const float*)d_in[19];
  P.bs_w1  = (const float*)d_in[20]; P.bs_b1  = (const float*)d_in[21];
  P.bs_w2  = (const float*)d_in[22]; P.bs_b2  = (const float*)d_in[23];
  P.out    = (float*)d_out;

  dim3 grid(BATCH / BLOCK);
  dim3 block(BLOCK);
  fsrs_scan<<<grid, block, 0, stream>>>(P);
}